// SoftDTWLoss_67688684585009
// MI455X (gfx1250) — compile-verified
//
#include <hip/hip_runtime.h>
#include <hip/hip_bf16.h>

typedef __attribute__((ext_vector_type(2))) float v2f;
typedef __attribute__((ext_vector_type(8))) float v8f;

#define TT 1024
#define CC 64
#define NB 16
#define GAMMA_F 0.01f
#define BIG_F 1e10f

// exp(-(a-m)/g) = exp2((m-a) * (1/g) * log2(e))
#define KEXP2 144.26950408889634f           // 100 * log2(e)
// g * ln(s) = (g*ln2) * log2(s)
#define KLOG2 0.0069314718055994529f        // 0.01 * ln(2)

// Raw hardware transcendentals: v_exp_f32 computes 2^x, v_log_f32 computes log2(x).
#define EXP2F(x) __builtin_amdgcn_exp2f(x)
#define LOG2F(x) __builtin_amdgcn_logf(x)

// ---------------------------------------------------------------------------
// Kernel 1: cost[b] = x2 + y2 - 2*x@y^T via fp32 WMMA (V_WMMA_F32_16X16X4_F32),
// stored SKEWED:  cost_skew[((i+j) mod 1024)*1024 + i]   (a bijection on
// [0,1024)^2) so DP anti-diagonals become contiguous 4KB runs. The 64MB
// buffer fits in the 192MB L2, so the scattered 4B stores coalesce there.
// One wave computes one 16x16 tile; x2/y2 folded in via an augmented K-step.
// ---------------------------------------------------------------------------
__global__ __launch_bounds__(32) void sdtw_cost_kernel(
    const float* __restrict__ x, const float* __restrict__ y,
    float* __restrict__ cost) {
  const int J = blockIdx.x;          // column tile 0..63
  const int I = blockIdx.y;          // row tile    0..63
  const int b = blockIdx.z;          // batch       0..15
  const int lane = threadIdx.x;      // 0..31 (wave32)
  const int l = lane & 15;
  const int half = lane >> 4;

  const float* xb = x + (size_t)b * TT * CC;
  const float* yb = y + (size_t)b * TT * CC;

  const int row = I * 16 + l;        // A-matrix row owned by this lane
  const int col = J * 16 + l;        // B-matrix col owned by this lane

  v8f acc = {};
  float x2p = 0.0f, y2p = 0.0f;      // per-half partial |x|^2, |y|^2

  // fp32 A-frag layout: VGPR0 = K{k0+0 | k0+2}, VGPR1 = K{k0+1 | k0+3} by
  // lane half -> contiguous float2 load at k0 + 2*half. B symmetric.
#pragma unroll
  for (int k0 = 0; k0 < CC; k0 += 4) {
    v2f a  = *(const v2f*)(xb + (size_t)row * CC + k0 + 2 * half);
    v2f yv = *(const v2f*)(yb + (size_t)col * CC + k0 + 2 * half);
    x2p += a.x * a.x + a.y * a.y;
    y2p += yv.x * yv.x + yv.y * yv.y;
    v2f bf;
    bf.x = -2.0f * yv.x;             // fold the -2 into B
    bf.y = -2.0f * yv.y;
    acc = __builtin_amdgcn_wmma_f32_16x16x4_f32(
        false, a, false, bf, (short)0, acc, false, false);
  }

  // Halves cover k%4 in {0,1} / {2,3}: combine across lane^16.
  float x2 = x2p + __shfl_xor(x2p, 16, 32);
  float y2 = y2p + __shfl_xor(y2p, 16, 32);

  // Augmented step K=[64..67]: A'=[x2, 1, 0, 0], B'=[1; y2; 0; 0]
  // contributes exactly x2[i] + y2[j] to every output element.
  v2f af, bf;
  if (half == 0) { af.x = x2;   af.y = 1.0f; bf.x = 1.0f; bf.y = y2; }
  else           { af.x = 0.0f; af.y = 0.0f; bf.x = 0.0f; bf.y = 0.0f; }
  acc = __builtin_amdgcn_wmma_f32_16x16x4_f32(
      false, af, false, bf, (short)0, acc, false, false);

  // C/D layout: VGPR v -> row M = v + 8*half, lanes hold N = l.
  float* cb = cost + (size_t)b * TT * TT;
#pragma unroll
  for (int v = 0; v < 8; ++v) {
    int i = I * 16 + v + 8 * half;
    int j = J * 16 + l;
    int dr = (i + j) & (TT - 1);
    cb[(size_t)dr * TT + i] = acc[v];
  }
}

// ---------------------------------------------------------------------------
// Kernel 2: anti-diagonal wavefront soft-DTW DP. One 1024-thread workgroup
// per batch; thread tid owns DP row i = tid+1. Three rotating LDS diagonals,
// one barrier per step. Cost cell for step d+1 is prefetched during step d
// (always in-bounds thanks to the skewed layout; invalid cells are masked),
// so the global-load latency overlaps compute + barrier instead of sitting
// on the 2047-step serial chain.
// ---------------------------------------------------------------------------
__global__ __launch_bounds__(1024) void sdtw_dp_kernel(
    const float* __restrict__ cost, float* __restrict__ result) {
  const int b = blockIdx.x;
  const int tid = threadIdx.x;
  const int i = tid + 1;             // DP row index 1..1024

  __shared__ float r[3][TT + 2];

  for (int t = tid; t <= TT; t += 1024) {
    r[0][t] = (t == 0) ? 0.0f : BIG_F;   // diagonal d=0
    r[1][t] = BIG_F;                     // diagonal d=1
  }
  __syncthreads();

  const float* cb = cost + (size_t)b * TT * TT;

  int p2 = 0, p1 = 1, pc = 2;
  float cur = BIG_F;

  // Prefetch cost for d=2: skew row (d-2)=0, col i-1.
  float cv = cb[(size_t)0 * TT + (i - 1)];

  for (int d = 2; d <= 2 * TT; ++d) {
    float cv_cur = cv;
    // Prefetch for step d+1: skew row (d-1)&1023 (in-bounds for all d,i).
    cv = cb[(size_t)((d - 1) & (TT - 1)) * TT + (i - 1)];

    float rp_i    = r[p1][i];        // R[i,   j-1]
    float rp_im1  = r[p1][i - 1];    // R[i-1, j  ]
    float rpp_im1 = r[p2][i - 1];    // R[i-1, j-1]

    int j = d - i;
    bool valid = (j >= 1) && (j <= TT);

    float m = fminf(fminf(rp_i, rp_im1), rpp_im1);
    float s = EXP2F((m - rp_i) * KEXP2)
            + EXP2F((m - rp_im1) * KEXP2)
            + EXP2F((m - rpp_im1) * KEXP2);
    cur = valid ? (cv_cur + m - KLOG2 * LOG2F(s)) : BIG_F;

    r[pc][i] = cur;
    if (tid == 0) r[pc][0] = BIG_F;

    int t = p2; p2 = p1; p1 = pc; pc = t;
    __syncthreads();
  }

  // Final cell R[T,T] is computed by tid = T-1 on diagonal d = 2T.
  if (tid == TT - 1) result[b] = cur;
}

// ---------------------------------------------------------------------------
// Kernel 3: deterministic sum of 16 per-batch distances.
// ---------------------------------------------------------------------------
__global__ void sdtw_reduce_kernel(const float* __restrict__ result,
                                   float* __restrict__ out) {
  if (threadIdx.x == 0) {
    float s = 0.0f;
    for (int b = 0; b < NB; ++b) s += result[b];
    out[0] = s;
  }
}

extern "C" void kernel_launch(void* const* d_in, const int* in_sizes, int n_in,
                              void* d_out, int out_size, void* d_ws, size_t ws_size,
                              hipStream_t stream) {
  const float* x = (const float*)d_in[0];   // (16,1024,64) f32
  const float* y = (const float*)d_in[1];   // (16,1024,64) f32

  float* cost    = (float*)d_ws;                                   // 64 MB skewed cost
  float* partial = (float*)((char*)d_ws +
                            (size_t)NB * TT * TT * sizeof(float)); // 16 floats

  dim3 gcost(TT / 16, TT / 16, NB);   // 64 x 64 tiles x 16 batches, 1 wave each
  sdtw_cost_kernel<<<gcost, 32, 0, stream>>>(x, y, cost);

  sdtw_dp_kernel<<<NB, 1024, 0, stream>>>(cost, partial);

  sdtw_reduce_kernel<<<1, 32, 0, stream>>>(partial, (float*)d_out);
}